// PatchletsExtractor_7078106104296
// MI455X (gfx1250) — compile-verified
//
#include <hip/hip_runtime.h>

// CDNA5 WMMA f32 16x16x4:  D[16x16] = A[16x4] * B[4x16] + C
// Trick: A row m = (qx, qy, qz, 1), B col n = (-2px, -2py, -2pz, |p|^2)
//        => D[m][n] = |p_n|^2 - 2 q_m . p_n   (squared distance minus |q|^2)
typedef float v2f __attribute__((ext_vector_type(2)));
typedef float v8f __attribute__((ext_vector_type(8)));

namespace {
constexpr int NPTS    = 1024;
constexpr int KNN     = 16;
constexpr int TSTEPS  = 32;
constexpr int BATCH   = 8;
constexpr int WAVES   = 4;
constexpr int QPW     = 32;              // queries per wave (rows 0..31 via two WMMAs)
constexpr int QPB     = WAVES * QPW;     // 128 queries per block
constexpr int CHUNK   = 64;              // points per LDS-staged chunk
constexpr int NCHUNK  = NPTS / CHUNK;    // 16
constexpr int RSTRIDE = CHUNK + 4;       // 68 floats: float4-aligned row, padded vs bank conflicts
}

__device__ __forceinline__ float lane_bcast(float v, int srcLane) {
  return __int_as_float(__builtin_amdgcn_ds_bpermute(srcLane << 2, __float_as_int(v)));
}

__global__ __launch_bounds__(WAVES * 32)
void patchlet_step_kernel(const float* __restrict__ pseq, float* out, int t)
{
  __shared__ __attribute__((aligned(16))) float sdist[WAVES][QPW][RSTRIDE];

  const int tid  = threadIdx.x;
  const int w    = tid >> 5;
  const int lane = tid & 31;
  const int half = lane >> 4;        // 0: K=0,1 slots ; 1: K=2,3 slots (A/B layout)
  const int lx   = lane & 15;
  const int b    = blockIdx.y;
  const int q    = blockIdx.x * QPB + w * QPW + lane;   // this lane's query

  // ---- flat output layout: (idxs, dist, idxs, pts, patchlet_feats, normalized, outx)
  const size_t S1 = (size_t)BATCH * TSTEPS * NPTS * KNN;   // 4,194,304
  const size_t S3 = S1 * 3;
  float* o_idx  = out;
  float* o_dist = out + S1;
  float* o_idx2 = out + 2 * S1;
  float* o_pts  = out + 3 * S1;
  float* o_pf   = out + 3 * S1 + S3;
  float* o_nrm  = out + 3 * S1 + 3 * S3;
  float* o_outx = out + 3 * S1 + 4 * S3;

  const float* P  = pseq + ((size_t)b * TSTEPS + t) * NPTS * 3;        // x_next
  const int tprev = (t > 0) ? (t - 1) : 0;
  const float* Fp = pseq + ((size_t)b * TSTEPS + tprev) * NPTS * 3;    // feat_prev

  // ---- load this lane's query: t==0 -> point_seq[:,0]; else nearest pt of previous step (outx[t-1])
  float qx, qy, qz;
  if (t == 0) {
    const float* Q = pseq + ((size_t)b * TSTEPS * NPTS + q) * 3;
    qx = Q[0]; qy = Q[1]; qz = Q[2];
  } else {
    const float* Q = o_outx + (((size_t)b * TSTEPS + (t - 1)) * NPTS + q) * 3;
    qx = Q[0]; qy = Q[1]; qz = Q[2];
  }
  const float q2 = qx * qx + qy * qy + qz * qz;

  // ---- pack A operands (16x4 f32 layout: lanes0-15 hold K0/K1, lanes16-31 hold K2/K3)
  const float xl = lane_bcast(qx, lx),        yl = lane_bcast(qy, lx),        zl = lane_bcast(qz, lx);
  const float xh = lane_bcast(qx, lx | 16),   yh = lane_bcast(qy, lx | 16),   zh = lane_bcast(qz, lx | 16);
  v2f a0, a1;                                  // a0: query rows 0..15, a1: rows 16..31
  a0[0] = half ? zl : xl;   a0[1] = half ? 1.0f : yl;
  a1[0] = half ? zh : xh;   a1[1] = half ? 1.0f : yh;

  // ---- top-16 running list (registers, sorted ascending; strict < keeps lowest-index ties first)
  float bd[KNN]; int bi[KNN];
#pragma unroll
  for (int i = 0; i < KNN; ++i) { bd[i] = 3.0e38f; bi[i] = 0; }

  auto consider = [&](float s, int jj) {
    if (s < bd[KNN - 1]) {
      bd[KNN - 1] = s; bi[KNN - 1] = jj;
#pragma unroll
      for (int i = KNN - 1; i > 0; --i) {
        if (bd[i] < bd[i - 1]) {
          float td = bd[i]; bd[i] = bd[i - 1]; bd[i - 1] = td;
          int   ti = bi[i]; bi[i] = bi[i - 1]; bi[i - 1] = ti;
        }
      }
    }
  };

  for (int c = 0; c < NCHUNK; ++c) {
    // ---- distance tiles via WMMA, scattered to wave-private LDS rows
#pragma unroll
    for (int tt = 0; tt < CHUNK / 16; ++tt) {
      const int pi = c * CHUNK + tt * 16 + lx;
      const float* pp = P + (size_t)pi * 3;
      const float px = pp[0], py = pp[1], pz = pp[2];
      const float p2 = px * px + py * py + pz * pz;
      v2f bv;                                  // B 4x16 layout mirrors A: half0 K0/K1, half1 K2/K3
      bv[0] = half ? (-2.0f * pz) : (-2.0f * px);
      bv[1] = half ? p2           : (-2.0f * py);
      v8f cz = {0.f, 0.f, 0.f, 0.f, 0.f, 0.f, 0.f, 0.f};
      v8f d0 = __builtin_amdgcn_wmma_f32_16x16x4_f32(false, a0, false, bv, (short)0, cz, false, false);
      v8f d1 = __builtin_amdgcn_wmma_f32_16x16x4_f32(false, a1, false, bv, (short)0, cz, false, false);
      const int col   = tt * 16 + lx;
      const int rbase = half * 8;              // C/D layout: vgpr r = row r (lanes0-15), r+8 (lanes16-31)
#pragma unroll
      for (int r = 0; r < 8; ++r) {
        sdist[w][rbase + r][col]      = d0[r];
        sdist[w][16 + rbase + r][col] = d1[r];
      }
    }
    // ---- each lane scans its own query's row (wave-private LDS: no barrier needed)
    const float4* row = reinterpret_cast<const float4*>(&sdist[w][lane][0]);
    for (int j4 = 0; j4 < CHUNK / 4; ++j4) {
      float4 v = row[j4];
      const int jb = c * CHUNK + j4 * 4;
      consider(v.x, jb + 0);
      consider(v.y, jb + 1);
      consider(v.z, jb + 2);
      consider(v.w, jb + 3);
    }
  }

  // ---- emit results for this query
  const size_t base_k  = (((size_t)b * TSTEPS + t) * NPTS + q) * KNN;
  const size_t base_k3 = base_k * 3;
  const size_t base_k6 = base_k * 6;

  const float* p0 = P + (size_t)bi[0] * 3;     // nearest point = new query
  const float nx = p0[0], ny = p0[1], nz = p0[2];

  float ax, ay, az;                            // anchor = nearest point at t=0
  if (t == 0) { ax = nx; ay = ny; az = nz; }
  else {
    const float* A = o_outx + ((size_t)b * TSTEPS * NPTS + q) * 3;   // outx[b,0,q] (written by t=0 launch)
    ax = A[0]; ay = A[1]; az = A[2];
  }

  {
    float* ox = o_outx + (((size_t)b * TSTEPS + t) * NPTS + q) * 3;
    ox[0] = nx; ox[1] = ny; ox[2] = nz;
  }

#pragma unroll
  for (int i = 0; i < KNN; ++i) {
    const int idx = bi[i];
    const float fid = (float)idx;
    o_idx [base_k + i] = fid;
    o_idx2[base_k + i] = fid;
    o_dist[base_k + i] = bd[i] + q2;           // add back |q|^2
    const float* pp = P  + (size_t)idx * 3;
    const float px = pp[0], py = pp[1], pz = pp[2];
    const float* fp = Fp + (size_t)idx * 3;
    o_pts[base_k3 + i * 3 + 0] = px;
    o_pts[base_k3 + i * 3 + 1] = py;
    o_pts[base_k3 + i * 3 + 2] = pz;
    const float rx = px - ax, ry = py - ay, rz = pz - az;
    o_nrm[base_k3 + i * 3 + 0] = rx;
    o_nrm[base_k3 + i * 3 + 1] = ry;
    o_nrm[base_k3 + i * 3 + 2] = rz;
    o_pf[base_k6 + i * 6 + 0] = fp[0];
    o_pf[base_k6 + i * 6 + 1] = fp[1];
    o_pf[base_k6 + i * 6 + 2] = fp[2];
    o_pf[base_k6 + i * 6 + 3] = rx;
    o_pf[base_k6 + i * 6 + 4] = ry;
    o_pf[base_k6 + i * 6 + 5] = rz;
  }
}

extern "C" void kernel_launch(void* const* d_in, const int* in_sizes, int n_in,
                              void* d_out, int out_size, void* d_ws, size_t ws_size,
                              hipStream_t stream) {
  (void)in_sizes; (void)n_in; (void)d_ws; (void)ws_size; (void)out_size;
  const float* pseq = (const float*)d_in[0];
  float* out = (float*)d_out;
  dim3 grid(NPTS / QPB, BATCH);   // 8 query-blocks x 8 batches
  // scan over t is sequentially dependent (queries at t come from outx[t-1]):
  // 32 stream-ordered launches (graph-capture safe).
  for (int t = 0; t < TSTEPS; ++t) {
    patchlet_step_kernel<<<grid, WAVES * 32, 0, stream>>>(pseq, out, t);
  }
}